// DynamicFilter_88545045774416
// MI455X (gfx1250) — compile-verified
//
#include <hip/hip_runtime.h>
#include <hip/hip_bf16.h>

// ---------------- problem constants ----------------
#define Bn   16
#define Hn   56
#define Wn   56
#define DIMn 384
#define MEDn 768            // 2*DIM
#define NFn  4
#define RHn  96
#define WFn  29             // W/2+1
#define EPSn 1e-5f
#define NPIX (Bn*Hn*Wn)     // 50176

typedef __attribute__((ext_vector_type(2))) float v2f;
typedef __attribute__((ext_vector_type(8))) float v8f;

#if defined(__has_builtin)
#if __has_builtin(__builtin_amdgcn_global_load_async_to_lds_b128)
#define HAS_ASYNC_LDS 1
#endif
#endif

// ==================================================================
// WMMA f32 GEMM:  C[M,N] = A[M,K] * B[K,N]   (row major)
// block tile 128x64, BK=16; 8 waves, each wave: 64 rows x 16 cols
// Double-buffered async global->LDS staging (ASYNCcnt) when available.
// epi=1 -> StarReLU(scale,bias) epilogue
// ==================================================================
#define BM 128
#define BN 64
#define BK 16

#if HAS_ASYNC_LDS
// builtin signature (from compiler diagnostic): pointee is a 16-byte int vector
typedef int v4i_vs __attribute__((vector_size(4 * sizeof(int))));
typedef __attribute__((address_space(1))) v4i_vs* gv4p;   // global (__device__)
typedef __attribute__((address_space(3))) v4i_vs* lv4p;   // LDS (__shared__)

// Stage one 128x16 A tile + 16x64 B tile: 768 16-byte chunks, 3 per thread.
// Each wave issues exactly 3 async-to-LDS instructions (ASYNCcnt += 3).
__device__ __forceinline__ void stage_tile_async(
    const float* A, const float* Bm, int K, int N,
    int row0, int col0, int k0, int tid,
    float (*sAb)[BK + 1], float (*sBb)[BN])
{
    #pragma unroll
    for (int j = 0; j < 3; ++j) {
        int chunk = tid + j * 256;          // uniform per wave (32-chunk blocks)
        if (chunk < 512) {                  // A tile: 512 float4 chunks
            int r  = chunk >> 2;
            int c4 = (chunk & 3) << 2;
            const float* gp = A + (size_t)(row0 + r) * K + k0 + c4;
            __builtin_amdgcn_global_load_async_to_lds_b128(
                (gv4p)gp, (lv4p)&sAb[r][c4], 0, 0);
        } else {                            // B tile: 256 float4 chunks
            int cb = chunk - 512;
            int r  = cb >> 4;
            int c4 = (cb & 15) << 2;
            const float* gp = Bm + (size_t)(k0 + r) * N + col0 + c4;
            __builtin_amdgcn_global_load_async_to_lds_b128(
                (gv4p)gp, (lv4p)&sBb[r][c4], 0, 0);
        }
    }
}
#endif

// One BK=16 step: preload all fragments (burst of ds_loads, single wait),
// then 16 back-to-back v_wmma_f32_16x16x4_f32.
__device__ __forceinline__ void compute_tile(
    const float (*cA)[BK + 1], const float (*cB)[BN],
    int wr, int wc, int lidx, int lhalf, v8f* acc)
{
    v2f bfr[4];
    v2f afr[4][4];
    #pragma unroll
    for (int q = 0; q < 4; ++q) {
        const int ks = q * 4 + lhalf * 2;
        bfr[q][0] = cB[ks    ][wc * 16 + lidx];
        bfr[q][1] = cB[ks + 1][wc * 16 + lidx];
        #pragma unroll
        for (int rt = 0; rt < 4; ++rt) {
            const int r = wr * 64 + rt * 16 + lidx;
            afr[q][rt][0] = cA[r][ks];
            afr[q][rt][1] = cA[r][ks + 1];
        }
    }
    #pragma unroll
    for (int q = 0; q < 4; ++q)
        #pragma unroll
        for (int rt = 0; rt < 4; ++rt)
            acc[rt] = __builtin_amdgcn_wmma_f32_16x16x4_f32(
                false, afr[q][rt], false, bfr[q], (short)0, acc[rt], false, false);
}

__global__ __launch_bounds__(256)
void wmma_gemm_f32(const float* __restrict__ A, const float* __restrict__ Bm,
                   float* __restrict__ C, int M, int N, int K,
                   const float* __restrict__ scale, const float* __restrict__ bias,
                   int epi)
{
    const int tid   = threadIdx.x;
    const int row0  = blockIdx.y * BM;
    const int col0  = blockIdx.x * BN;
    const int wave  = tid >> 5;
    const int lane  = tid & 31;
    const int wr    = wave >> 2;        // 0..1  (64-row half)
    const int wc    = wave & 3;         // 0..3  (16-col strip)
    const int lidx  = lane & 15;
    const int lhalf = lane >> 4;

    v8f acc[4];
    #pragma unroll
    for (int t = 0; t < 4; ++t)
        #pragma unroll
        for (int e = 0; e < 8; ++e) acc[t][e] = 0.0f;

#if HAS_ASYNC_LDS
    __shared__ float sA[2][BM][BK + 1];
    __shared__ float sB[2][BK][BN];
    const int nt = K / BK;
    stage_tile_async(A, Bm, K, N, row0, col0, 0, tid, sA[0], sB[0]);
    for (int t = 0; t < nt; ++t) {
        const int cur = t & 1;
        if (t + 1 < nt) {
            stage_tile_async(A, Bm, K, N, row0, col0, (t + 1) * BK, tid,
                             sA[cur ^ 1], sB[cur ^ 1]);
            // 3 outstanding for next tile; wait until current tile's 3 land
            asm volatile("s_wait_asynccnt 0x3" ::: "memory");
        } else {
            asm volatile("s_wait_asynccnt 0x0" ::: "memory");
        }
        __syncthreads();                       // tile t visible to all waves
        compute_tile(sA[cur], sB[cur], wr, wc, lidx, lhalf, acc);
        __syncthreads();                       // done reading before overwrite
    }
#else
    __shared__ float sA[BM][BK + 1];
    __shared__ float sB[BK][BN];
    for (int k0 = 0; k0 < K; k0 += BK) {
        __syncthreads();
        #pragma unroll
        for (int it = 0; it < 2; ++it) {
            int idx4 = tid + it * 256;
            int r  = idx4 >> 2;
            int c4 = (idx4 & 3) << 2;
            const float4 va = *(const float4*)(A + (size_t)(row0 + r) * K + k0 + c4);
            sA[r][c4 + 0] = va.x; sA[r][c4 + 1] = va.y;
            sA[r][c4 + 2] = va.z; sA[r][c4 + 3] = va.w;
        }
        {
            int r  = tid >> 4;
            int c4 = (tid & 15) << 2;
            const float4 vb = *(const float4*)(Bm + (size_t)(k0 + r) * N + col0 + c4);
            *(float4*)&sB[r][c4] = vb;
        }
        __syncthreads();
        compute_tile(sA, sB, wr, wc, lidx, lhalf, acc);
    }
#endif

    float s = 0.f, bb = 0.f;
    if (epi) { s = scale[0]; bb = bias[0]; }
    #pragma unroll
    for (int rt = 0; rt < 4; ++rt) {
        #pragma unroll
        for (int e = 0; e < 8; ++e) {
            int gr = row0 + wr * 64 + rt * 16 + lhalf * 8 + e;
            int gc = col0 + wc * 16 + lidx;
            float val = acc[rt][e];
            if (epi) { float r = fmaxf(val, 0.f); val = s * r * r + bb; }
            C[(size_t)gr * N + gc] = val;
        }
    }
}

// ==================================================================
// small helpers
// ==================================================================
__global__ __launch_bounds__(256)
void zero_kernel(float* p, int n)
{
    int i = blockIdx.x * 256 + threadIdx.x;
    if (i < n) p[i] = 0.f;
}

// global average pool: g[b,d] = mean over H*W
__global__ __launch_bounds__(256)
void gap_kernel(const float* __restrict__ x, float* __restrict__ g)
{
    int i = blockIdx.x * 256 + threadIdx.x;      // i < B*DIM
    if (i >= Bn * DIMn) return;
    int b = i / DIMn, d = i % DIMn;
    const float* p = x + (size_t)b * Hn * Wn * DIMn + d;
    float ssum = 0.f;
    for (int t = 0; t < Hn * Wn; ++t) ssum += p[(size_t)t * DIMn];
    g[i] = ssum * (1.0f / (Hn * Wn));
}

// routing MLP + softmax over NF; one block per batch element
__global__ __launch_bounds__(256)
void routing_kernel(const float* __restrict__ g,
                    const float* __restrict__ w_r1,
                    const float* __restrict__ r_scale, const float* __restrict__ r_bias,
                    const float* __restrict__ w_r2,
                    float* __restrict__ routing /*[B][NF][MED]*/)
{
    __shared__ float hsh[RHn];
    const int b = blockIdx.x, t = threadIdx.x;
    if (t < RHn) {
        float a = 0.f;
        for (int d = 0; d < DIMn; ++d) a += g[b * DIMn + d] * w_r1[d * RHn + t];
        float r = fmaxf(a, 0.f);
        hsh[t] = r_scale[0] * r * r + r_bias[0];
    }
    __syncthreads();
    for (int m = t; m < MEDn; m += 256) {
        float l[NFn];
        #pragma unroll
        for (int f = 0; f < NFn; ++f) {
            float a = 0.f;
            for (int r = 0; r < RHn; ++r)
                a += hsh[r] * w_r2[r * (NFn * MEDn) + f * MEDn + m];
            l[f] = a;
        }
        float mx = fmaxf(fmaxf(l[0], l[1]), fmaxf(l[2], l[3]));
        float e[NFn], ssum = 0.f;
        #pragma unroll
        for (int f = 0; f < NFn; ++f) { e[f] = expf(l[f] - mx); ssum += e[f]; }
        float inv = 1.0f / ssum;
        #pragma unroll
        for (int f = 0; f < NFn; ++f)
            routing[((size_t)b * NFn + f) * MEDn + m] = e[f] * inv;
    }
}

// depthwise 3x3 SAME conv + bias -> loc_raw
__global__ __launch_bounds__(256)
void dwconv_kernel(const float* __restrict__ v, const float* __restrict__ kern,
                   const float* __restrict__ dwb, float* __restrict__ loc)
{
    size_t i = (size_t)blockIdx.x * 256 + threadIdx.x;  // B*H*W*MED
    int c = (int)(i % MEDn);
    size_t p = i / MEDn;
    int w = (int)(p % Wn); size_t q = p / Wn;
    int h = (int)(q % Hn); int b = (int)(q / Hn);
    float acc = dwb[c];
    #pragma unroll
    for (int dy = 0; dy < 3; ++dy) {
        int hh = h + dy - 1;
        if (hh < 0 || hh >= Hn) continue;
        #pragma unroll
        for (int dx = 0; dx < 3; ++dx) {
            int ww = w + dx - 1;
            if (ww < 0 || ww >= Wn) continue;
            acc += v[(((size_t)b * Hn + hh) * Wn + ww) * MEDn + c]
                 * kern[(dy * 3 + dx) * MEDn + c];
        }
    }
    loc[i] = acc;
}

// per-channel sum/sumsq over B*H*W  (128 blocks, each covers 392 pixels)
__global__ __launch_bounds__(256)
void stats_kernel(const float* __restrict__ loc,
                  float* __restrict__ sum, float* __restrict__ sumsq)
{
    const int PPB = NPIX / 128;
    size_t p0 = (size_t)blockIdx.x * PPB;
    float as[3] = {0.f, 0.f, 0.f}, aq[3] = {0.f, 0.f, 0.f};
    for (int pp = 0; pp < PPB; ++pp) {
        const float* row = loc + (p0 + pp) * (size_t)MEDn;
        #pragma unroll
        for (int j = 0; j < 3; ++j) {
            float vv = row[threadIdx.x + j * 256];
            as[j] += vv; aq[j] += vv * vv;
        }
    }
    #pragma unroll
    for (int j = 0; j < 3; ++j) {
        atomicAdd(&sum[threadIdx.x + j * 256], as[j]);
        atomicAdd(&sumsq[threadIdx.x + j * 256], aq[j]);
    }
}

__global__ __launch_bounds__(256)
void bnfin_kernel(const float* __restrict__ sum, const float* __restrict__ sumsq,
                  const float* __restrict__ gamma, const float* __restrict__ beta,
                  float* __restrict__ bnsc, float* __restrict__ bnsh)
{
    int c = blockIdx.x * 256 + threadIdx.x;
    if (c >= MEDn) return;
    const float invN = 1.0f / (float)NPIX;
    float mu  = sum[c] * invN;
    float var = sumsq[c] * invN - mu * mu;
    float sc  = gamma[c] * rsqrtf(var + EPSn);
    bnsc[c] = sc;
    bnsh[c] = beta[c] - mu * sc;
}

// ==================================================================
// spectral branch: separable direct DFTs (56 is small)
// ==================================================================
#define TWOPI 6.283185307179586f

// rfft along W:  T1[b,h,k,c] = (1/56) * sum_w v[b,h,w,c] e^{-2pi i k w/56}
__global__ __launch_bounds__(256)
void rfft_w_kernel(const float* __restrict__ v,
                   float* __restrict__ T1re, float* __restrict__ T1im)
{
    __shared__ float twc[56], tws[56];
    if (threadIdx.x < 56) {
        float ang = TWOPI * (float)threadIdx.x / 56.0f;
        twc[threadIdx.x] = cosf(ang);
        tws[threadIdx.x] = sinf(ang);
    }
    __syncthreads();
    size_t i = (size_t)blockIdx.x * 256 + threadIdx.x;   // B*H*WF*MED
    int c = (int)(i % MEDn);
    size_t p = i / MEDn;
    int k = (int)(p % WFn); size_t q = p / WFn;
    int h = (int)(q % Hn);  int b = (int)(q / Hn);
    const float* src = v + ((size_t)(b * Hn + h) * Wn) * MEDn + c;
    float sr = 0.f, si = 0.f;
    int idx = 0;
    for (int w = 0; w < Wn; ++w) {
        float vv = src[(size_t)w * MEDn];
        sr += vv * twc[idx];
        si -= vv * tws[idx];
        idx += k; if (idx >= 56) idx -= 56;
    }
    T1re[i] = sr * (1.0f / 56.0f);
    T1im[i] = si * (1.0f / 56.0f);
}

// DFT along H (forward) fused with dynamic complex filter:
// X[b,u,k,c] = w[b,u,k,c] * sum_h T1[b,h,k,c] e^{-2pi i u h/56}
__global__ __launch_bounds__(256)
void hfwd_kernel(const float* __restrict__ T1re, const float* __restrict__ T1im,
                 const float* __restrict__ routing, const float* __restrict__ cw,
                 float* __restrict__ Xre, float* __restrict__ Xim)
{
    __shared__ float twc[56], tws[56];
    if (threadIdx.x < 56) {
        float ang = TWOPI * (float)threadIdx.x / 56.0f;
        twc[threadIdx.x] = cosf(ang);
        tws[threadIdx.x] = sinf(ang);
    }
    __syncthreads();
    size_t i = (size_t)blockIdx.x * 256 + threadIdx.x;   // B*H*WF*MED
    int c = (int)(i % MEDn);
    size_t p = i / MEDn;
    int k = (int)(p % WFn); size_t q = p / WFn;
    int u = (int)(q % Hn);  int b = (int)(q / Hn);
    size_t base = (((size_t)b * Hn) * WFn + k) * MEDn + c;
    const size_t hs = (size_t)WFn * MEDn;
    float Xr = 0.f, Xi = 0.f;
    int idx = 0;
    for (int h = 0; h < Hn; ++h) {
        if (h + 8 < Hn)
            __builtin_prefetch(&T1re[base + (size_t)(h + 8) * hs], 0, 1);
        float tr = T1re[base + (size_t)h * hs];
        float ti = T1im[base + (size_t)h * hs];
        float cc = twc[idx], ss = tws[idx];     // e^{-i} = cc - i ss
        Xr += tr * cc + ti * ss;
        Xi += ti * cc - tr * ss;
        idx += u; if (idx >= 56) idx -= 56;
    }
    const float* cwp = cw + ((size_t)u * WFn + k) * (NFn * 2);
    float r0 = routing[((size_t)b * NFn + 0) * MEDn + c];
    float r1 = routing[((size_t)b * NFn + 1) * MEDn + c];
    float r2 = routing[((size_t)b * NFn + 2) * MEDn + c];
    float r3 = routing[((size_t)b * NFn + 3) * MEDn + c];
    float wr = r0 * cwp[0] + r1 * cwp[2] + r2 * cwp[4] + r3 * cwp[6];
    float wi = r0 * cwp[1] + r1 * cwp[3] + r2 * cwp[5] + r3 * cwp[7];
    Xre[i] = Xr * wr - Xi * wi;
    Xim[i] = Xr * wi + Xi * wr;
}

// inverse DFT along H: Y[b,hh,k,c] = sum_u X[b,u,k,c] e^{+2pi i u hh/56}
__global__ __launch_bounds__(256)
void hinv_kernel(const float* __restrict__ Xre, const float* __restrict__ Xim,
                 float* __restrict__ Yre, float* __restrict__ Yim)
{
    __shared__ float twc[56], tws[56];
    if (threadIdx.x < 56) {
        float ang = TWOPI * (float)threadIdx.x / 56.0f;
        twc[threadIdx.x] = cosf(ang);
        tws[threadIdx.x] = sinf(ang);
    }
    __syncthreads();
    size_t i = (size_t)blockIdx.x * 256 + threadIdx.x;   // B*H*WF*MED
    int c = (int)(i % MEDn);
    size_t p = i / MEDn;
    int k = (int)(p % WFn); size_t q = p / WFn;
    int hh = (int)(q % Hn); int b = (int)(q / Hn);
    size_t base = (((size_t)b * Hn) * WFn + k) * MEDn + c;
    const size_t hs = (size_t)WFn * MEDn;
    float Yr = 0.f, Yi = 0.f;
    int idx = 0;
    for (int u = 0; u < Hn; ++u) {
        if (u + 8 < Hn)
            __builtin_prefetch(&Xre[base + (size_t)(u + 8) * hs], 0, 1);
        float xr = Xre[base + (size_t)u * hs];
        float xi = Xim[base + (size_t)u * hs];
        float cc = twc[idx], ss = tws[idx];     // e^{+i} = cc + i ss
        Yr += xr * cc - xi * ss;
        Yi += xr * ss + xi * cc;
        idx += hh; if (idx >= 56) idx -= 56;
    }
    Yre[i] = Yr;
    Yim[i] = Yi;
}

// irfft along W (Hermitian half -> 56 reals, 1/56 ortho factor),
// fused with BN-affine + StarReLU of the local branch and residual add.
__global__ __launch_bounds__(256)
void irfft_combine_kernel(const float* __restrict__ Yre, const float* __restrict__ Yim,
                          const float* __restrict__ loc,
                          const float* __restrict__ bnsc, const float* __restrict__ bnsh,
                          const float* __restrict__ l_scale, const float* __restrict__ l_bias,
                          float* __restrict__ ucomb)
{
    __shared__ float twc[56], tws[56];
    if (threadIdx.x < 56) {
        float ang = TWOPI * (float)threadIdx.x / 56.0f;
        twc[threadIdx.x] = cosf(ang);
        tws[threadIdx.x] = sinf(ang);
    }
    __syncthreads();
    size_t i = (size_t)blockIdx.x * 256 + threadIdx.x;   // B*H*W*MED
    int c = (int)(i % MEDn);
    size_t p = i / MEDn;
    int w = (int)(p % Wn); size_t q = p / Wn;
    int h = (int)(q % Hn); int b = (int)(q / Hn);
    size_t base = ((size_t)(b * Hn + h) * WFn) * MEDn + c;
    float acc = 0.f;
    int idx = 0;
    for (int k = 0; k < WFn; ++k) {
        float yr = Yre[base + (size_t)k * MEDn];
        float yi = Yim[base + (size_t)k * MEDn];
        float term = yr * twc[idx] - yi * tws[idx];
        float alpha = (k == 0 || k == WFn - 1) ? 1.0f : 2.0f;
        acc += alpha * term;
        idx += w; if (idx >= 56) idx -= 56;
    }
    float y = acc * (1.0f / 56.0f);
    float lv = loc[i] * bnsc[c] + bnsh[c];
    float r = fmaxf(lv, 0.f);
    lv = l_scale[0] * r * r + l_bias[0];
    ucomb[i] = y + lv;
}

// ==================================================================
// launch
// ==================================================================
extern "C" void kernel_launch(void* const* d_in, const int* in_sizes, int n_in,
                              void* d_out, int out_size, void* d_ws, size_t ws_size,
                              hipStream_t stream)
{
    const float* x        = (const float*)d_in[0];
    const float* w_pw1    = (const float*)d_in[1];
    const float* w_pw2    = (const float*)d_in[2];
    const float* a1_scale = (const float*)d_in[3];
    const float* a1_bias  = (const float*)d_in[4];
    const float* w_r1     = (const float*)d_in[5];
    const float* r_scale  = (const float*)d_in[6];
    const float* r_bias   = (const float*)d_in[7];
    const float* w_r2     = (const float*)d_in[8];
    const float* dw_kern  = (const float*)d_in[9];
    const float* dw_bias  = (const float*)d_in[10];
    const float* bn_gamma = (const float*)d_in[11];
    const float* bn_beta  = (const float*)d_in[12];
    const float* l_scale  = (const float*)d_in[13];
    const float* l_bias   = (const float*)d_in[14];
    const float* cw       = (const float*)d_in[15];
    float* out = (float*)d_out;

    // workspace carve-up (floats)
    const size_t NV = (size_t)Bn * Hn * Wn * MEDn;   // 38,535,168
    const size_t NS = (size_t)Bn * Hn * WFn * MEDn;  // 19,955,712
    float* ws   = (float*)d_ws;
    float* v    = ws;                 // [B,H,W,MED]  (later reused for y+loc)
    float* loc  = v    + NV;          // [B,H,W,MED]
    float* T1re = loc  + NV;          // [B,H,WF,MED]
    float* T1im = T1re + NS;
    float* Xre  = T1im + NS;
    float* Xim  = Xre  + NS;
    float* g    = Xim  + NS;          // [B,DIM]
    float* rout = g    + (size_t)Bn * DIMn;          // [B,NF,MED]
    float* ssum = rout + (size_t)Bn * NFn * MEDn;    // [MED]
    float* ssq  = ssum + MEDn;                        // [MED]
    float* bnsc = ssq  + MEDn;                        // [MED]
    float* bnsh = bnsc + MEDn;                        // [MED]

    // 1) zero BN accumulators (must be re-zeroed every call)
    zero_kernel<<<dim3((2 * MEDn + 255) / 256), dim3(256), 0, stream>>>(ssum, 2 * MEDn);

    // 2) routing path
    gap_kernel<<<dim3((Bn * DIMn + 255) / 256), dim3(256), 0, stream>>>(x, g);
    routing_kernel<<<dim3(Bn), dim3(256), 0, stream>>>(g, w_r1, r_scale, r_bias, w_r2, rout);

    // 3) v = StarReLU(x @ w_pw1)   -- WMMA GEMM, M=50176 N=768 K=384
    {
        dim3 grid(MEDn / BN, (Bn * Hn * Wn) / BM);
        wmma_gemm_f32<<<grid, dim3(256), 0, stream>>>(
            x, w_pw1, v, Bn * Hn * Wn, MEDn, DIMn, a1_scale, a1_bias, 1);
    }

    // 4) local branch: depthwise conv, batch stats, BN coefficients
    dwconv_kernel<<<dim3((unsigned)(NV / 256)), dim3(256), 0, stream>>>(v, dw_kern, dw_bias, loc);
    stats_kernel<<<dim3(128), dim3(256), 0, stream>>>(loc, ssum, ssq);
    bnfin_kernel<<<dim3((MEDn + 255) / 256), dim3(256), 0, stream>>>(
        ssum, ssq, bn_gamma, bn_beta, bnsc, bnsh);

    // 5) spectral branch
    rfft_w_kernel<<<dim3((unsigned)(NS / 256)), dim3(256), 0, stream>>>(v, T1re, T1im);
    hfwd_kernel<<<dim3((unsigned)(NS / 256)), dim3(256), 0, stream>>>(
        T1re, T1im, rout, cw, Xre, Xim);
    hinv_kernel<<<dim3((unsigned)(NS / 256)), dim3(256), 0, stream>>>(Xre, Xim, T1re, T1im);

    // 6) irfft + BN/StarReLU(loc) + residual; write into v buffer (v is dead now)
    irfft_combine_kernel<<<dim3((unsigned)(NV / 256)), dim3(256), 0, stream>>>(
        T1re, T1im, loc, bnsc, bnsh, l_scale, l_bias, v);

    // 7) out = (y + loc) @ w_pw2   -- WMMA GEMM, M=50176 N=384 K=768
    {
        dim3 grid(DIMn / BN, (Bn * Hn * Wn) / BM);
        wmma_gemm_f32<<<grid, dim3(256), 0, stream>>>(
            v, w_pw2, out, Bn * Hn * Wn, DIMn, MEDn, (const float*)nullptr,
            (const float*)nullptr, 0);
    }
}